// MultiHeadSelfAttention_283467841554
// MI455X (gfx1250) — compile-verified
//
#include <hip/hip_runtime.h>

typedef __attribute__((ext_vector_type(2))) float v2f;
typedef __attribute__((ext_vector_type(8))) float v8f;
typedef __attribute__((ext_vector_type(4))) int   v4i;

#define B_  2
#define L_  2048
#define D_  1024
#define H_  16
#define HD_ 64
#define M_  (B_ * L_)

__device__ __forceinline__ v8f wmma4(v2f a, v2f b, v8f c) {
  // D = A(16x4 f32) * B(4x16 f32) + C(16x16 f32)
  return __builtin_amdgcn_wmma_f32_16x16x4_f32(
      /*neg_a=*/false, a, /*neg_b=*/false, b,
      /*c_mod=*/(short)0, c, /*reuse_a=*/false, /*reuse_b=*/false);
}

// ---- async global->LDS support (probe-gated) ------------------------------
#if __has_builtin(__builtin_amdgcn_global_load_async_to_lds_b128)
#define HAVE_ASYNC_LDS 1
typedef __attribute__((address_space(1))) v4i* g128p;
typedef __attribute__((address_space(3))) v4i* l128p;
__device__ __forceinline__ void async_ld128(const float* g, float* l) {
  __builtin_amdgcn_global_load_async_to_lds_b128((g128p)g, (l128p)l, 0, 0);
}
__device__ __forceinline__ void wait_async0() {
#if __has_builtin(__builtin_amdgcn_s_wait_asynccnt)
  __builtin_amdgcn_s_wait_asynccnt(0);
#else
  asm volatile("s_wait_asynccnt 0" ::: "memory");
#endif
}
#endif

// ---------------------------------------------------------------------------
// y = x @ W^T + bias   (x: [M,K] row-major, W: [N,K] row-major, y: [M,N])
// 128x64 tile / block, 8 waves (4x2), each wave owns a 32x32 slab
// (four 16x16 accumulators). Async path: double-buffered LDS, tile i+1
// streamed by the async-DMA pipe while tile i is consumed by WMMA.
// ---------------------------------------------------------------------------
#define GK 16
#define GS 20  // padded LDS row stride (floats): 16B-aligned rows, conflict-free
#define ABUF (128 * GS)
#define BBUF (64 * GS)

__global__ __launch_bounds__(256) void gemm_xwt_bias(
    const float* __restrict__ X, const float* __restrict__ W,
    const float* __restrict__ bias, float* __restrict__ Y,
    int Ndim, int Kdim) {
  __shared__ float As[2 * ABUF];
  __shared__ float Bs[2 * BBUF];

  const int tid  = threadIdx.x;
  const int lane = tid & 31;
  const int wave = tid >> 5;
  const int wm   = wave & 3;   // 32-row group
  const int wn   = wave >> 2;  // 32-col group
  const int half = lane >> 4;  // 0/1 -> K sub-pair for frags, row+8 for C
  const int l16  = lane & 15;

  const int arow = tid >> 1;          // 0..127 (A tile row)
  const int acol = (tid & 1) << 3;    // 0 or 8
  const int brow = tid >> 2;          // 0..63  (B tile row)
  const int bcol = (tid & 3) << 2;    // 0,4,8,12

  const int bm = blockIdx.x * 128;
  const int bn = blockIdx.y * 64;

  v8f c00 = {0.f, 0.f, 0.f, 0.f, 0.f, 0.f, 0.f, 0.f};
  v8f c01 = c00, c10 = c00, c11 = c00;

  const float* xrow = X + (size_t)(bm + arow) * Kdim + acol;
  const float* wrow = W + (size_t)(bn + brow) * Kdim + bcol;

#if HAVE_ASYNC_LDS
  // ---- async, double-buffered pipeline ----
  async_ld128(xrow,     &As[arow * GS + acol]);
  async_ld128(xrow + 4, &As[arow * GS + acol + 4]);
  async_ld128(wrow,     &Bs[brow * GS + bcol]);
  wait_async0();
  __syncthreads();

  int cur = 0;
  for (int k0 = 0; k0 < Kdim; k0 += GK) {
    const int nxt = cur ^ 1;
    if (k0 + GK < Kdim) {
      async_ld128(xrow + k0 + GK,     &As[nxt * ABUF + arow * GS + acol]);
      async_ld128(xrow + k0 + GK + 4, &As[nxt * ABUF + arow * GS + acol + 4]);
      async_ld128(wrow + k0 + GK,     &Bs[nxt * BBUF + brow * GS + bcol]);
    }
    const float* a0p = &As[cur * ABUF + (wm * 32 + l16) * GS + half * 2];
    const float* a1p = a0p + 16 * GS;
    const float* b0p = &Bs[cur * BBUF + (wn * 32 + l16) * GS + half * 2];
    const float* b1p = b0p + 16 * GS;
#pragma unroll
    for (int kk = 0; kk < GK; kk += 4) {
      v2f a0 = *(const v2f*)(a0p + kk);
      v2f a1 = *(const v2f*)(a1p + kk);
      v2f b0 = *(const v2f*)(b0p + kk);
      v2f b1 = *(const v2f*)(b1p + kk);
      c00 = wmma4(a0, b0, c00);
      c01 = wmma4(a0, b1, c01);
      c10 = wmma4(a1, b0, c10);
      c11 = wmma4(a1, b1, c11);
    }
    wait_async0();      // next tile resident in LDS (own wave's copies)
    __syncthreads();    // all waves done reading 'cur', all copies visible
    cur = nxt;
  }
#else
  // ---- fallback: VGPR-staged, single buffer ----
  for (int k0 = 0; k0 < Kdim; k0 += GK) {
    *(float4*)&As[arow * GS + acol]     = *(const float4*)(xrow + k0);
    *(float4*)&As[arow * GS + acol + 4] = *(const float4*)(xrow + k0 + 4);
    *(float4*)&Bs[brow * GS + bcol]     = *(const float4*)(wrow + k0);
    __syncthreads();

    const float* a0p = &As[(wm * 32 + l16) * GS + half * 2];
    const float* a1p = a0p + 16 * GS;
    const float* b0p = &Bs[(wn * 32 + l16) * GS + half * 2];
    const float* b1p = b0p + 16 * GS;
#pragma unroll
    for (int kk = 0; kk < GK; kk += 4) {
      v2f a0 = *(const v2f*)(a0p + kk);
      v2f a1 = *(const v2f*)(a1p + kk);
      v2f b0 = *(const v2f*)(b0p + kk);
      v2f b1 = *(const v2f*)(b1p + kk);
      c00 = wmma4(a0, b0, c00);
      c01 = wmma4(a0, b1, c01);
      c10 = wmma4(a1, b0, c10);
      c11 = wmma4(a1, b1, c11);
    }
    __syncthreads();
  }
#endif

  const int n0    = bn + wn * 32 + l16;
  const float bv0 = bias[n0];
  const float bv1 = bias[n0 + 16];
  const int r0    = bm + wm * 32 + half * 8;
#pragma unroll
  for (int v = 0; v < 8; ++v) {
    Y[(size_t)(r0 + v) * Ndim + n0]           = c00[v] + bv0;
    Y[(size_t)(r0 + v) * Ndim + n0 + 16]      = c01[v] + bv1;
    Y[(size_t)(r0 + 16 + v) * Ndim + n0]      = c10[v] + bv0;
    Y[(size_t)(r0 + 16 + v) * Ndim + n0 + 16] = c11[v] + bv1;
  }
}

// ---------------------------------------------------------------------------
// Fused flash-attention: one block per (batch, head, 64-query tile).
// Q/K/V layout: [B*L, D] row-major, head h occupies columns h*64..h*64+63.
// Online softmax parallelized 4 threads/row; score tile lives only in LDS.
// ---------------------------------------------------------------------------
#define AS 68  // padded stride (floats): rows 16B aligned, conflict-free wmma frags

__global__ __launch_bounds__(256) void attn_fused(
    const float* __restrict__ Q, const float* __restrict__ K,
    const float* __restrict__ V, const int* __restrict__ mask,
    float* __restrict__ O) {
  __shared__ float Ks[64 * AS];   // [key][hd]
  __shared__ float Vt[64 * AS];   // [hd][key]   (transposed)
  __shared__ float Ss[64 * AS];   // [q][key]    (also used to stage Q once)
  __shared__ float pmax[256];
  __shared__ float psum[256];
  __shared__ float rowf[64];
  __shared__ float rowinv[64];

  const int tid  = threadIdx.x;
  const int lane = tid & 31;
  const int wave = tid >> 5;
  const int wm   = wave & 3;
  const int wn   = wave >> 2;
  const int half = lane >> 4;
  const int l16  = lane & 15;
  const int lrow = tid >> 2;        // 0..63
  const int lcol = (tid & 3) << 4;  // 0,16,32,48

  const int q0 = blockIdx.x * 64;
  const int h  = blockIdx.y;
  const int b  = blockIdx.z;

  const size_t headoff = (size_t)h * HD_;
  const float* qbase   = Q + (size_t)(b * L_ + q0) * D_ + headoff;

  // stage Q tile through LDS (coalesced), then pull fragments to registers
#pragma unroll
  for (int j = 0; j < 16; j += 4)
    *(float4*)&Ss[lrow * AS + lcol + j] =
        *(const float4*)(qbase + (size_t)lrow * D_ + lcol + j);
  __syncthreads();

  v2f aq[16];
  {
    const float scale = 0.125f;  // 1/sqrt(HD)
    const float* qp = &Ss[(wm * 16 + l16) * AS + half * 2];
#pragma unroll
    for (int c = 0; c < 16; ++c) {
      v2f t = *(const v2f*)(qp + c * 4);
      aq[c] = t * scale;
    }
  }

  v8f o0 = {0.f, 0.f, 0.f, 0.f, 0.f, 0.f, 0.f, 0.f};
  v8f o1 = o0;
  float run_max = -1e30f;
  float run_sum = 0.f;

  for (int kt = 0; kt < L_ / 64; ++kt) {
    const int k0 = kt * 64;
    const float* kb = K + (size_t)(b * L_ + k0) * D_ + headoff;
    const float* vb = V + (size_t)(b * L_ + k0) * D_ + headoff;

    // load K tile row-major, V tile transposed
#pragma unroll
    for (int j = 0; j < 16; j += 4) {
      *(float4*)&Ks[lrow * AS + lcol + j] =
          *(const float4*)(kb + (size_t)lrow * D_ + lcol + j);
      float4 vv = *(const float4*)(vb + (size_t)lrow * D_ + lcol + j);
      Vt[(lcol + j + 0) * AS + lrow] = vv.x;
      Vt[(lcol + j + 1) * AS + lrow] = vv.y;
      Vt[(lcol + j + 2) * AS + lrow] = vv.z;
      Vt[(lcol + j + 3) * AS + lrow] = vv.w;
    }
    // prefetch next K/V tile into cache while we compute on this one
    if (kt + 1 < L_ / 64) {
      __builtin_prefetch(kb + 64 * D_ + (size_t)lrow * D_ + lcol, 0, 0);
      __builtin_prefetch(vb + 64 * D_ + (size_t)lrow * D_ + lcol, 0, 0);
    }
    __syncthreads();

    // S = (Q*scale) @ K^T : two 16x16 tiles per wave
    v8f s0 = {0.f, 0.f, 0.f, 0.f, 0.f, 0.f, 0.f, 0.f};
    v8f s1 = s0;
    {
      const float* kp0 = &Ks[(wn * 32 + l16) * AS + half * 2];
      const float* kp1 = &Ks[(wn * 32 + 16 + l16) * AS + half * 2];
#pragma unroll
      for (int c = 0; c < 16; ++c) {
        v2f b0 = *(const v2f*)(kp0 + c * 4);
        v2f b1 = *(const v2f*)(kp1 + c * 4);
        s0 = wmma4(aq[c], b0, s0);
        s1 = wmma4(aq[c], b1, s1);
      }
    }
    // spill scores to LDS (conflict-free: bank = 4*row + col)
    {
      float* sp = &Ss[(wm * 16 + half * 8) * AS + wn * 32 + l16];
#pragma unroll
      for (int v = 0; v < 8; ++v) {
        sp[v * AS]      = s0[v];
        sp[v * AS + 16] = s1[v];
      }
    }
    __syncthreads();

    // online softmax: 4 threads per q row, 16 columns each.
    {
      const int r   = tid >> 2;
      const int seg = tid & 3;
      float* srow   = &Ss[r * AS + seg * 16];
      const int* mrow = mask + b * L_ + k0 + seg * 16;

      float bmax = -1e30f;
#pragma unroll
      for (int j = 0; j < 16; ++j) {
        float s = srow[j];
        if (mrow[j] == 0) s = 1e-9f;  // faithful to reference
        srow[j] = s;
        bmax = fmaxf(bmax, s);
      }
      pmax[tid] = bmax;
      __syncthreads();

      bmax = fmaxf(fmaxf(pmax[r * 4 + 0], pmax[r * 4 + 1]),
                   fmaxf(pmax[r * 4 + 2], pmax[r * 4 + 3]));
      float nm = fmaxf(run_max, bmax);
      float f  = __expf(run_max - nm);
      if (seg == 0) rowf[r] = f;

      float ls = 0.f;
#pragma unroll
      for (int j = 0; j < 16; ++j) {
        float e = __expf(srow[j] - nm);
        srow[j] = e;
        ls += e;
      }
      psum[tid] = ls;
      __syncthreads();

      float bs = psum[r * 4 + 0] + psum[r * 4 + 1] +
                 psum[r * 4 + 2] + psum[r * 4 + 3];
      run_sum = run_sum * f + bs;
      run_max = nm;
    }

    // rescale accumulators by exp(old_max - new_max)
    {
      const float* fp = &rowf[wm * 16 + half * 8];
#pragma unroll
      for (int v = 0; v < 8; ++v) {
        float f = fp[v];
        o0[v] *= f;
        o1[v] *= f;
      }
    }

    // O += P @ V
    {
      const float* pp  = &Ss[(wm * 16 + l16) * AS + half * 2];
      const float* vp0 = &Vt[(wn * 32 + l16) * AS + half * 2];
      const float* vp1 = &Vt[(wn * 32 + 16 + l16) * AS + half * 2];
#pragma unroll
      for (int c = 0; c < 16; ++c) {
        v2f a  = *(const v2f*)(pp + c * 4);
        v2f b0 = *(const v2f*)(vp0 + c * 4);
        v2f b1 = *(const v2f*)(vp1 + c * 4);
        o0 = wmma4(a, b0, o0);
        o1 = wmma4(a, b1, o1);
      }
    }
    __syncthreads();
  }

  if (tid < 64) rowinv[tid] = 1.0f / run_sum;
  __syncthreads();

  {
    const float* ip = &rowinv[wm * 16 + half * 8];
    float* obase = O + (size_t)(b * L_ + q0 + wm * 16 + half * 8) * D_ +
                   headoff + wn * 32 + l16;
#pragma unroll
    for (int v = 0; v < 8; ++v) {
      float inv = ip[v];
      obase[(size_t)v * D_]      = o0[v] * inv;
      obase[(size_t)v * D_ + 16] = o1[v] * inv;
    }
  }
}

// ---------------------------------------------------------------------------
extern "C" void kernel_launch(void* const* d_in, const int* in_sizes, int n_in,
                              void* d_out, int out_size, void* d_ws, size_t ws_size,
                              hipStream_t stream) {
  (void)in_sizes; (void)n_in; (void)out_size; (void)ws_size;
  const float* x    = (const float*)d_in[0];
  const int*   mask = (const int*)d_in[1];
  const float* Wq   = (const float*)d_in[2];
  const float* bq   = (const float*)d_in[3];
  const float* Wk   = (const float*)d_in[4];
  const float* bk   = (const float*)d_in[5];
  const float* Wv   = (const float*)d_in[6];
  const float* bv   = (const float*)d_in[7];
  const float* Wo   = (const float*)d_in[8];
  const float* bo   = (const float*)d_in[9];
  float* out = (float*)d_out;
  float* ws  = (float*)d_ws;

  const size_t MD = (size_t)M_ * D_;
  float* Qb = ws;
  float* Kb = ws + MD;
  float* Vb = ws + 2 * MD;
  float* Ab = ws + 3 * MD;

  dim3 gg(M_ / 128, D_ / 64);
  gemm_xwt_bias<<<gg, 256, 0, stream>>>(x, Wq, bq, Qb, D_, D_);
  gemm_xwt_bias<<<gg, 256, 0, stream>>>(x, Wk, bk, Kb, D_, D_);
  gemm_xwt_bias<<<gg, 256, 0, stream>>>(x, Wv, bv, Vb, D_, D_);
  attn_fused<<<dim3(L_ / 64, H_, B_), 256, 0, stream>>>(Qb, Kb, Vb, mask, Ab);
  gemm_xwt_bias<<<gg, 256, 0, stream>>>(Ab, Wo, bo, out, D_, D_);
}